// GraphConv_72610717106653
// MI455X (gfx1250) — compile-verified
//
#include <hip/hip_runtime.h>

typedef __attribute__((ext_vector_type(2))) float v2f;
typedef __attribute__((ext_vector_type(8))) float v8f;

#define DIN   256
#define DOUT  64
#define XSTR  (DIN + 4)   // padded LDS row stride (floats): 260 -> conflict-free A-frag reads

// ---------------------------------------------------------------------------
// Kernel 1: support = x @ W + b   via V_WMMA_F32_16X16X4_F32
//   block = 256 threads (8 waves); block computes 32 rows x 64 cols
//   wave w: mt = w>>2 (0..1), nt = w&3 (0..3) -> one 16x16 tile per wave
//
// LDS layout forced via struct: the x tile is at offset 0 so A-fragment
// ds_load byte offsets (row-base in VGPR + k*4 <= 1020) fit the 16-bit DS
// immediate field (and pair into ds_load_2addr_b64). The W buffer sits above;
// its B-fragment loads use a VGPR base + 8-bit x512B 2addr_stride64 offsets,
// so a high base costs one add total, not one per load.
// ---------------------------------------------------------------------------
struct GemmSmem {
  float  x[32 * XSTR];          // ~32.5 KB @ offset 0: 32-row x tile, padded rows
  float2 w[(DIN / 2) * DOUT];   // 64 KB: wlds[kp*64+n]=(W[2kp][n],W[2kp+1][n])
};

__global__ __launch_bounds__(256) void gc_gemm_wmma(const float* __restrict__ x,
                                                    const float* __restrict__ W,
                                                    const float* __restrict__ bias,
                                                    float* __restrict__ support,
                                                    int nnodes) {
  __shared__ GemmSmem sm;

  const int tid     = threadIdx.x;
  const int lane    = tid & 31;
  const int wave    = tid >> 5;
  const int rowBase = blockIdx.x * 32;

  // ---- stage W (16384 floats) into LDS in (k-pair, n) float2 layout ----
  #pragma unroll
  for (int i = 0; i < ((DIN / 2) * DOUT) / 256; ++i) {
    int idx = tid + i * 256;            // float2 index: 0..8191
    int kp  = idx >> 6;                 // k-pair 0..127
    int n   = idx & 63;                 // col 0..63
    sm.w[idx] = make_float2(W[(kp * 2 + 0) * DOUT + n],
                            W[(kp * 2 + 1) * DOUT + n]);
  }

  // ---- stage x tile (32 x 256 floats) into padded LDS, fully coalesced ----
  {
    const float4* xg = (const float4*)x;
    #pragma unroll
    for (int i = 0; i < 8; ++i) {
      int idx  = tid + i * 256;         // float4 index within tile: 0..2047
      int gr   = idx >> 6;              // tile row 0..31
      int gc   = idx & 63;              // float4 col 0..63
      int grow = rowBase + gr;
      if (grow > nnodes - 1) grow = nnodes - 1;   // clamp (nnodes%32==0 normally)
      *(float4*)&sm.x[gr * XSTR + (gc << 2)] = xg[(size_t)grow * (DIN / 4) + gc];
    }
  }
  __syncthreads();

  const int nt   = wave & 3;
  const int mt   = wave >> 2;
  const int ncol = nt * 16 + (lane & 15);          // output column 0..63
  const int kOff = (lane >> 4) * 2;                // lanes 0-15: K={0,1}; 16-31: K={2,3}

  // A fragment source: row (mt*16 + lane&15), starting at kOff (array @ LDS 0)
  const float*  arow = &sm.x[(mt * 16 + (lane & 15)) * XSTR + kOff];
  // B fragment source: k-pair offset (lane>>4) baked into base
  const float2* bcol = &sm.w[(lane >> 4) * DOUT + ncol];

  v8f c0 = {};
  v8f c1 = {};
  #pragma unroll
  for (int k0 = 0; k0 < DIN; k0 += 8) {
    // chain 0: K = k0..k0+3
    v2f a0; a0.x = arow[k0];     a0.y = arow[k0 + 1];
    float2 w0 = bcol[(k0 >> 1) * DOUT];
    v2f b0; b0.x = w0.x;         b0.y = w0.y;
    c0 = __builtin_amdgcn_wmma_f32_16x16x4_f32(false, a0, false, b0,
                                               (short)0, c0, false, false);
    // chain 1: K = k0+4..k0+7 (independent accumulator for ILP)
    v2f a1; a1.x = arow[k0 + 4]; a1.y = arow[k0 + 5];
    float2 w1 = bcol[((k0 + 4) >> 1) * DOUT];
    v2f b1; b1.x = w1.x;         b1.y = w1.y;
    c1 = __builtin_amdgcn_wmma_f32_16x16x4_f32(false, a1, false, b1,
                                               (short)0, c1, false, false);
  }

  // D layout (32-bit C/D 16x16): lanes 0-15 -> M = vgpr, lanes 16-31 -> M = 8+vgpr
  const float bv    = bias[ncol];
  const int   mBase = rowBase + mt * 16 + (lane >> 4) * 8;
  #pragma unroll
  for (int i = 0; i < 8; ++i) {
    int m = mBase + i;
    if (m < nnodes)
      support[(size_t)m * DOUT + ncol] = c0[i] + c1[i] + bv;
  }
}

// ---------------------------------------------------------------------------
// Kernel 2: zero the output (harness poisons d_out; must re-init every call)
// ---------------------------------------------------------------------------
__global__ __launch_bounds__(256) void gc_zero4(float4* __restrict__ out, int n4) {
  int i = blockIdx.x * 256 + threadIdx.x;
  if (i < n4) out[i] = make_float4(0.f, 0.f, 0.f, 0.f);
}

// ---------------------------------------------------------------------------
// Kernel 3: SpMM scatter:  out[dst[e]] += val[e] * support[src[e]]
//   one wave32 per edge; lane owns features {lane, lane+32}
//   support & out both fit in the 192MB L2 -> atomics resolve at L2 speed
// ---------------------------------------------------------------------------
__global__ __launch_bounds__(256) void gc_spmm(const int*   __restrict__ esrc,
                                               const int*   __restrict__ edst,
                                               const float* __restrict__ eval,
                                               const float* __restrict__ support,
                                               float*       __restrict__ out,
                                               int nedges) {
  const int lane = threadIdx.x & 31;
  // edge id is wave-uniform; force scalar so edge metadata uses the scalar path
  const int e = __builtin_amdgcn_readfirstlane(blockIdx.x * 8 + (threadIdx.x >> 5));
  if (e >= nedges) return;

  const int   s = esrc[e];
  const int   d = edst[e];
  const float v = eval[e];

  const float* sp = support + (size_t)s * DOUT;
  float m0 = sp[lane]      * v;
  float m1 = sp[lane + 32] * v;

  float* op = out + (size_t)d * DOUT;
  atomicAdd(op + lane,      m0);
  atomicAdd(op + lane + 32, m1);
}

// ---------------------------------------------------------------------------
extern "C" void kernel_launch(void* const* d_in, const int* in_sizes, int n_in,
                              void* d_out, int out_size, void* d_ws, size_t ws_size,
                              hipStream_t stream) {
  const float* x    = (const float*)d_in[0];
  const int*   esrc = (const int*)  d_in[1];
  const int*   edst = (const int*)  d_in[2];
  const float* eval = (const float*)d_in[3];
  const float* W    = (const float*)d_in[4];
  const float* bias = (const float*)d_in[5];
  float*       out  = (float*)d_out;
  float*       supp = (float*)d_ws;          // [nnodes, 64] scratch

  const int nnodes = in_sizes[0] / DIN;      // 100000
  const int nedges = in_sizes[1];            // 1600000

  // zero output (poisoned by harness)
  const int n4 = (nnodes * DOUT) / 4;
  gc_zero4<<<(n4 + 255) / 256, 256, 0, stream>>>((float4*)out, n4);

  // dense projection with f32 WMMA
  gc_gemm_wmma<<<(nnodes + 31) / 32, 256, 0, stream>>>(x, W, bias, supp, nnodes);

  // sparse aggregation (1 wave per edge, 8 waves per block)
  gc_spmm<<<(nedges + 7) / 8, 256, 0, stream>>>(esrc, edst, eval, supp, out, nedges);
}